// ContrastiveLoss_59287728554067
// MI455X (gfx1250) — compile-verified
//
#include <hip/hip_runtime.h>

// Math: loss = [(N-1)(Sx+Sy) - 2*(sx.sy) + 2*Dxy] / (2*N*(N-1))
//   Sx = sum ||x_i||^2, Sy = sum ||y_j||^2, sx/sy = column sums, Dxy = sum_i x_i.y_i
// Everything is a streaming reduction over 8 MB -> HBM-bound (~0.35 us @ 23.3 TB/s).
// Reductions are done on the matrix pipe: V_WMMA_F32_16X16X4_F32 with an all-ones
// A-matrix sums a 4x16 fp32 tile per issue (D[m][n] = sum_k B[k][n]).

typedef __attribute__((ext_vector_type(2))) float v2f;
typedef __attribute__((ext_vector_type(8))) float v8f;

#define DIM   128
#define GRID  64
#define WAVES 8
#define BLOCK (WAVES * 32)

__global__ __launch_bounds__(BLOCK) void cl_partials(const float* __restrict__ X,
                                                     const float* __restrict__ Y,
                                                     float* __restrict__ ws,
                                                     int n)
{
    const int wave = threadIdx.x >> 5;   // wave32
    const int lane = threadIdx.x & 31;
    const int half = lane >> 4;          // which K-pair this lane carries
    const int col  = (wave << 4) + (lane & 15);   // this wave owns 16 columns

    const int rowsPerBlock = n / GRID;            // 128 for N=8192
    const int rowStart = blockIdx.x * rowsPerBlock;
    const int rowEnd   = rowStart + rowsPerBlock;

    const v2f ones = {1.0f, 1.0f};   // A-matrix = all ones -> WMMA does column sums
    v8f cx  = {};  // colsum X
    v8f cy  = {};  // colsum Y
    v8f cx2 = {};  // colsum X*X
    v8f cy2 = {};  // colsum Y*Y
    v8f cxy = {};  // colsum X*Y (diagonal dot)

    for (int r = rowStart; r < rowEnd; r += 4) {
        // B-matrix (4x16): lanes 0-15 carry rows r+0 (b.x) and r+1 (b.y),
        // lanes 16-31 carry rows r+2 / r+3. Since A is all-ones, any bijective
        // K assignment yields the correct column sum.
        const int r0 = r + (half << 1);
        const int r1 = r0 + 1;
        v2f xb, yb;
        xb.x = X[r0 * DIM + col];
        xb.y = X[r1 * DIM + col];
        yb.x = Y[r0 * DIM + col];
        yb.y = Y[r1 * DIM + col];
        v2f x2 = xb * xb;
        v2f y2 = yb * yb;
        v2f xy = xb * yb;

        cx  = __builtin_amdgcn_wmma_f32_16x16x4_f32(false, ones, false, xb, (short)0, cx,  false, false);
        cy  = __builtin_amdgcn_wmma_f32_16x16x4_f32(false, ones, false, yb, (short)0, cy,  false, false);
        cx2 = __builtin_amdgcn_wmma_f32_16x16x4_f32(false, ones, false, x2, (short)0, cx2, false, false);
        cy2 = __builtin_amdgcn_wmma_f32_16x16x4_f32(false, ones, false, y2, (short)0, cy2, false, false);
        cxy = __builtin_amdgcn_wmma_f32_16x16x4_f32(false, ones, false, xy, (short)0, cxy, false, false);
    }

    // D row M=0 lives in VGPR0 of lanes 0-15 (column n = lane): all rows of D
    // are identical colsums, so one lane per column writes out.
    if (lane < 16) {
        float* base = ws + (size_t)blockIdx.x * 5 * DIM;
        base[0 * DIM + col] = cx[0];
        base[1 * DIM + col] = cy[0];
        base[2 * DIM + col] = cx2[0];
        base[3 * DIM + col] = cy2[0];
        base[4 * DIM + col] = cxy[0];
    }
}

__global__ __launch_bounds__(DIM) void cl_final(const float* __restrict__ ws,
                                                float* __restrict__ out,
                                                int n)
{
    __shared__ float sm[4][DIM];
    const int d = threadIdx.x;   // one thread per feature dimension

    float sx = 0.f, sy = 0.f, x2 = 0.f, y2 = 0.f, xy = 0.f;
    for (int b = 0; b < GRID; ++b) {
        const float* base = ws + (size_t)b * 5 * DIM;
        sx += base[0 * DIM + d];
        sy += base[1 * DIM + d];
        x2 += base[2 * DIM + d];
        y2 += base[3 * DIM + d];
        xy += base[4 * DIM + d];
    }

    sm[0][d] = x2;
    sm[1][d] = y2;
    sm[2][d] = xy;
    sm[3][d] = sx * sy;   // contribution to sx . sy
    __syncthreads();

    for (int s = DIM / 2; s > 0; s >>= 1) {
        if (d < s) {
            sm[0][d] += sm[0][d + s];
            sm[1][d] += sm[1][d + s];
            sm[2][d] += sm[2][d + s];
            sm[3][d] += sm[3][d + s];
        }
        __syncthreads();
    }

    if (d == 0) {
        const double Nd  = (double)n;
        const double Sx  = sm[0][0];
        const double Sy  = sm[1][0];
        const double Dxy = sm[2][0];
        const double P   = sm[3][0];
        const double num = (Nd - 1.0) * (Sx + Sy) - 2.0 * P + 2.0 * Dxy;
        out[0] = (float)(num / (2.0 * Nd * (Nd - 1.0)));
    }
}

extern "C" void kernel_launch(void* const* d_in, const int* in_sizes, int n_in,
                              void* d_out, int out_size, void* d_ws, size_t ws_size,
                              hipStream_t stream) {
    const float* X = (const float*)d_in[0];   // feature1 [N,128] f32
    const float* Y = (const float*)d_in[1];   // feature2 [N,128] f32
    // d_in[2] (label) is mathematically unused by the reference result.
    const int n = in_sizes[0] / DIM;          // 8192

    float* ws  = (float*)d_ws;                // 64 * 5 * 128 floats = 160 KB
    float* out = (float*)d_out;

    cl_partials<<<GRID, BLOCK, 0, stream>>>(X, Y, ws, n);
    cl_final<<<1, DIM, 0, stream>>>(ws, out, n);
}